// NRILayer_91164975825448
// MI455X (gfx1250) — compile-verified
//
#include <hip/hip_runtime.h>
#include <math.h>

typedef __attribute__((ext_vector_type(2))) float v2f;
typedef __attribute__((ext_vector_type(4))) float v4f;
typedef __attribute__((ext_vector_type(8))) float v8f;

#define BDIM 4
#define NDIM 256
#define HDIM 128
#define EPSL 1e-5f
#define AS_STRIDE 132   // 16-row A tile, padded: bank-conflict-free ds_load_b64
#define S_STRIDE  132

__device__ __forceinline__ v8f wmma_f32_4(v2f a, v2f b, v8f c) {
  // V_WMMA_F32_16X16X4_F32 : D = A(16x4) * B(4x16) + C(16x16)
  return __builtin_amdgcn_wmma_f32_16x16x4_f32(
      /*neg_a=*/false, a, /*neg_b=*/false, b,
      /*c_mod=*/(short)0, c, /*reuse_a=*/false, /*reuse_b=*/false);
}

__device__ __forceinline__ float wave_sum32(float x) {
  #pragma unroll
  for (int off = 16; off > 0; off >>= 1) x += __shfl_xor(x, off, 32);
  return x;
}

__device__ __forceinline__ float elu1(float x) {
  return x > 0.f ? x : (__expf(x) - 1.f);
}

__device__ __forceinline__ void wait_async0() {
#if __has_builtin(__builtin_amdgcn_s_wait_asynccnt)
  __builtin_amdgcn_s_wait_asynccnt(0);
#else
  asm volatile("s_wait_asynccnt 0x0" ::: "memory");
#endif
}

// Async copy 32B (2 x b128) from global to LDS; INST_OFFSET applies to both
// the LDS address and the global
// address, so one base pair covers both halves.
__device__ __forceinline__ void async_copy32B(unsigned int lds_byte_addr,
                                              const float* gsrc) {
  asm volatile(
      "global_load_async_to_lds_b128 %0, %1, off\n\t"
      "global_load_async_to_lds_b128 %0, %1, off offset:16"
      :: "v"(lds_byte_addr), "v"(gsrc)
      : "memory");
}

// ---------------------------------------------------------------- zero agg
__global__ void k_zero(float* __restrict__ p, int n) {
  int i = blockIdx.x * blockDim.x + threadIdx.x;
  if (i < n) p[i] = 0.f;
}

// ------------------------------------------------- vin / vout projections
// one wave per 16x16 output tile; blockIdx.z selects W_in vs W_out
__global__ __launch_bounds__(32) void k_proj(
    const float* __restrict__ v,
    const float* __restrict__ Win, const float* __restrict__ bin,
    const float* __restrict__ Wout, const float* __restrict__ bout,
    float* __restrict__ vin, float* __restrict__ vout)
{
  const int lane = threadIdx.x & 31;
  const int lh = lane & 15, lq = lane >> 4;
  const int ct = blockIdx.x;            // H tile 0..7
  const int rt = blockIdx.y;            // row tile 0..63
  const int which = blockIdx.z;         // 0: in, 1: out
  const float* W    = which ? Wout : Win;
  const float* bias = which ? bout : bin;
  float*       dst  = which ? vout : vin;
  const int r0 = rt * 16, h0 = ct * 16;

  const float* Ap = v + (r0 + lh) * HDIM + 2 * lq;   // A[m][k] fragment base
  const float* Bp = W + (h0 + lh) * HDIM + 2 * lq;   // B[k][n] = W[n][k]
  v8f c = {};
  #pragma unroll
  for (int kk = 0; kk < 32; ++kk) {
    v2f a  = *(const v2f*)(Ap + 4 * kk);
    v2f bb = *(const v2f*)(Bp + 4 * kk);
    c = wmma_f32_4(a, bb, c);
  }
  const float bv = bias[h0 + lh];
  #pragma unroll
  for (int g = 0; g < 8; ++g)
    dst[(r0 + g + 8 * lq) * HDIM + h0 + lh] = c[g] + bv;
}

// -------------------------------------------------------- fused edge pass
// block = 256 thr (8 waves). block -> (b, j-tile of 16, i-chunk of 16).
// wave w owns output columns h in [16w, 16w+16).
__global__ __launch_bounds__(256) void k_edge(
    const float* __restrict__ e,    const float* __restrict__ vin,
    const float* __restrict__ vout, const float* __restrict__ We2e,
    const float* __restrict__ be2e, const float* __restrict__ g_e,
    const float* __restrict__ be_e, const unsigned char* __restrict__ emask,
    float* __restrict__ eout,       float* __restrict__ agg)
{
  __shared__ float As[16 * AS_STRIDE];   // staged e tile (A matrix)
  __shared__ float Sln[16 * S_STRIDE];   // pre-LN activations
  __shared__ float aggS[16 * HDIM];      // per-block partial agg over i

  const int tid  = threadIdx.x;
  const int wave = tid >> 5, lane = tid & 31;
  const int lh = lane & 15, lq = lane >> 4;
  const int jt = blockIdx.x, ig = blockIdx.y, b = blockIdx.z;
  const int j0 = jt * 16, h0 = wave * 16;
  const int hcol = h0 + lh;

  // Preload all 32 B fragments of W_e2e^T for this wave's column tile.
  v2f bfrag[32];
  {
    const float* Bp = We2e + (h0 + lh) * HDIM + 2 * lq;  // B[k][n] = W[n][k]
    #pragma unroll
    for (int kk = 0; kk < 32; ++kk) bfrag[kk] = *(const v2f*)(Bp + 4 * kk);
  }
  // vout[b, j0+m, h] + b_e2e[h], constant over the i loop (m = g + 8*lq)
  float voutr[8];
  #pragma unroll
  for (int g = 0; g < 8; ++g)
    voutr[g] = vout[(b * NDIM + j0 + g + 8 * lq) * HDIM + hcol] + be2e[hcol];

  for (int idx = tid; idx < 16 * HDIM; idx += 256) aggS[idx] = 0.f;
  __syncthreads();

  const int arow = tid >> 4;          // staging: thread copies 8 floats (32B)
  const int acol = (tid & 15) * 8;
  const unsigned int lds_dst =
      (unsigned int)(size_t)(As + arow * AS_STRIDE + acol);

  for (int ii = 0; ii < 16; ++ii) {
    const int i = ig * 16 + ii;
    // e[b][i][j0..j0+15][:] is one contiguous 8KB chunk
    const float* src = e + (((size_t)(b * NDIM + i)) * NDIM + j0) * HDIM;
    // async memory -> LDS staging (GLOBAL_LOAD_ASYNC_TO_LDS_B128, ASYNCcnt)
    async_copy32B(lds_dst, src + arow * HDIM + acol);
    if (ii + 1 < 16)
      __builtin_prefetch(src + NDIM * HDIM + tid * 8, 0, 0);  // next i's tile
    wait_async0();
    __syncthreads();

    // 16x16 output tile per wave, K = 128 in 32 WMMA steps
    v8f c = {};
    {
      const float* Aprow = As + lh * AS_STRIDE + 2 * lq;
      #pragma unroll
      for (int kk = 0; kk < 32; ++kk) {
        v2f a = *(const v2f*)(Aprow + 4 * kk);
        c = wmma_f32_4(a, bfrag[kk], c);
      }
    }
    const float vins = vin[(b * NDIM + i) * HDIM + hcol];
    #pragma unroll
    for (int g = 0; g < 8; ++g)
      Sln[(g + 8 * lq) * S_STRIDE + hcol] = c[g] + vins + voutr[g];
    __syncthreads();

    // LayerNorm + ELU + masks; wave w handles rows m = w and w+8
    #pragma unroll
    for (int rr = 0; rr < 2; ++rr) {
      const int m = wave + rr * 8;
      const float* row = Sln + m * S_STRIDE;
      float x[4], s = 0.f, sq = 0.f;
      #pragma unroll
      for (int q = 0; q < 4; ++q) {
        x[q] = row[lane + 32 * q];
        s += x[q]; sq += x[q] * x[q];
      }
      s  = wave_sum32(s);
      sq = wave_sum32(sq);
      const float mean = s * (1.f / HDIM);
      const float var  = sq * (1.f / HDIM) - mean * mean;
      const float inv  = rsqrtf(var + EPSL);
      const int j = j0 + m;
      const bool kill = (i == j) ||
          (emask[((size_t)b * NDIM + i) * NDIM + j] != 0);
      float* orow = eout + (((size_t)(b * NDIM + i)) * NDIM + j) * HDIM;
      #pragma unroll
      for (int q = 0; q < 4; ++q) {
        const int h = lane + 32 * q;
        float y = (x[q] - mean) * inv * g_e[h] + be_e[h];
        y = elu1(y);
        if (kill) y = 0.f;
        orow[h] = y;                    // coalesced 128B per quarter-row
        aggS[m * HDIM + h] += y;        // unique (m,h) per thread: no race
      }
    }
    __syncthreads();
  }

  // flush partial agg (sum over this block's 16 i's) with f32 atomics
  for (int idx = tid; idx < 16 * HDIM; idx += 256) {
    const int m = idx >> 7, h = idx & 127;
    atomicAdd(&agg[(b * NDIM + j0 + m) * HDIM + h], aggS[idx]);
  }
}

// ------------------------------------------------------- node finalize
__global__ __launch_bounds__(128) void k_vnode(
    const float* __restrict__ v, const unsigned char* __restrict__ vmask,
    const float* __restrict__ agg, const float* __restrict__ g_v,
    const float* __restrict__ be_v, float* __restrict__ vnew)
{
  __shared__ float red[8];
  const int row = blockIdx.x;           // b*N + n
  const int t = threadIdx.x;            // h
  const int wave = t >> 5, lane = t & 31;
  const float x = agg[row * HDIM + t];
  float s = wave_sum32(x), sq = wave_sum32(x * x);
  if (lane == 0) { red[wave] = s; red[4 + wave] = sq; }
  __syncthreads();
  const float st  = red[0] + red[1] + red[2] + red[3];
  const float sqt = red[4] + red[5] + red[6] + red[7];
  const float mean = st * (1.f / HDIM);
  const float var  = sqt * (1.f / HDIM) - mean * mean;
  const float inv  = rsqrtf(var + EPSL);
  float y = (x - mean) * inv * g_v[t] + be_v[t];
  y = elu1(y);
  const float vm = vmask[row] ? 0.f : v[row * HDIM + t];
  vnew[row * HDIM + t] = vm + y;
}

// ---------------------------------------------------------------- launch
extern "C" void kernel_launch(void* const* d_in, const int* in_sizes, int n_in,
                              void* d_out, int out_size, void* d_ws, size_t ws_size,
                              hipStream_t stream) {
  const float* v    = (const float*)d_in[0];
  const float* e    = (const float*)d_in[1];
  const unsigned char* vmask = (const unsigned char*)d_in[2];
  const unsigned char* emask = (const unsigned char*)d_in[3];
  const float* Win  = (const float*)d_in[4];
  const float* bin  = (const float*)d_in[5];
  const float* Wout = (const float*)d_in[6];
  const float* bout = (const float*)d_in[7];
  const float* We2e = (const float*)d_in[8];
  const float* be2e = (const float*)d_in[9];
  const float* g_e  = (const float*)d_in[10];
  const float* be_e = (const float*)d_in[11];
  const float* g_v  = (const float*)d_in[12];
  const float* be_v = (const float*)d_in[13];

  float* out  = (float*)d_out;
  float* vnew = out;                          // [B,N,H]
  float* eout = out + BDIM * NDIM * HDIM;     // [B,N,N,H]

  float* ws   = (float*)d_ws;
  float* vin  = ws;                           // [B,N,H]
  float* vout = ws + BDIM * NDIM * HDIM;      // [B,N,H]
  float* agg  = ws + 2 * BDIM * NDIM * HDIM;  // [B,N,H]

  const int vElems = BDIM * NDIM * HDIM;
  k_zero<<<dim3((vElems + 255) / 256), 256, 0, stream>>>(agg, vElems);
  k_proj<<<dim3(HDIM / 16, BDIM * NDIM / 16, 2), 32, 0, stream>>>(
      v, Win, bin, Wout, bout, vin, vout);
  k_edge<<<dim3(NDIM / 16, NDIM / 16, BDIM), 256, 0, stream>>>(
      e, vin, vout, We2e, be2e, g_e, be_e, emask, eout, agg);
  k_vnode<<<dim3(BDIM * NDIM), 128, 0, stream>>>(
      v, vmask, agg, g_v, be_v, vnew);
}